// GGNN_86526411145929
// MI455X (gfx1250) — compile-verified
//
#include <hip/hip_runtime.h>

// Problem constants (from reference)
#define BB   32
#define NN   128
#define FF   10
#define DD   64
#define TOPK 20
#define EXTRA 25
#define BN   4096          // BB*NN
#define TWOBN 8192
#define EPS  1e-5f
#define SLOPE 0.2f

typedef __attribute__((ext_vector_type(2))) float v2f;
typedef __attribute__((ext_vector_type(8))) float v8f;

__device__ __forceinline__ float lrelu(float x) { return x > 0.f ? x : SLOPE * x; }

// monotone float<->uint mapping for atomicMax on floats (incl. negatives)
__device__ __forceinline__ unsigned fenc(float f) {
    unsigned u = __float_as_uint(f);
    return (u & 0x80000000u) ? ~u : (u | 0x80000000u);
}
__device__ __forceinline__ float fdec(unsigned u) {
    unsigned v = (u & 0x80000000u) ? (u & 0x7FFFFFFFu) : ~u;
    return __uint_as_float(v);
}

// ---------------------------------------------------------------------------
// Stage 1: cosine(emb, emb) top-20 per row (128x128, trivial). 1 block.
// ---------------------------------------------------------------------------
__global__ void k_topk_emb(const float* __restrict__ emb,
                           float* __restrict__ topv, int* __restrict__ topi) {
    __shared__ float invn[NN];
    int i = threadIdx.x;            // 128 threads, one row each
    float ei[DD];
    float ss = 0.f;
    for (int k = 0; k < DD; ++k) { ei[k] = emb[i * DD + k]; ss += ei[k] * ei[k]; }
    invn[i] = 1.0f / fmaxf(sqrtf(ss), 1e-12f);
    __syncthreads();
    float tv[TOPK]; int ti[TOPK];
    for (int k = 0; k < TOPK; ++k) { tv[k] = -3e38f; ti[k] = -1; }
    float inv_i = invn[i];
    for (int j = 0; j < NN; ++j) {
        const float* ej = emb + j * DD;
        float dot = 0.f;
        for (int k = 0; k < DD; ++k) dot += ei[k] * ej[k];
        float c = dot * inv_i * invn[j];
        if (c > tv[TOPK - 1]) {           // strict > keeps lax.top_k tie order
            int p = TOPK - 1;
            while (p > 0 && c > tv[p - 1]) { tv[p] = tv[p - 1]; ti[p] = ti[p - 1]; --p; }
            tv[p] = c; ti[p] = j;
        }
    }
    for (int k = 0; k < TOPK; ++k) { topv[i * TOPK + k] = tv[k]; topi[i * TOPK + k] = ti[k]; }
}

// ---------------------------------------------------------------------------
// h = x @ W1  (4096x10 @ 10x64) — tiny, scalar
// ---------------------------------------------------------------------------
__global__ void k_h(const float* __restrict__ x, const float* __restrict__ W1,
                    float* __restrict__ h) {
    int gid = blockIdx.x * blockDim.x + threadIdx.x;
    if (gid >= BN * DD) return;
    int r = gid >> 6, d = gid & 63;
    float a = 0.f;
    for (int f = 0; f < FF; ++f) a += x[r * FF + f] * W1[f * DD + d];
    h[gid] = a;
}

// si/sj per-node attention scalars: kf = [h, emb_tile] @ att
__global__ void k_sisj(const float* __restrict__ h, const float* __restrict__ emb,
                       const float* __restrict__ ai, const float* __restrict__ aj,
                       float* __restrict__ si, float* __restrict__ sj) {
    int r = blockIdx.x * blockDim.x + threadIdx.x;
    if (r >= BN) return;
    int n = r & (NN - 1);
    float a = 0.f, b = 0.f;
    for (int d = 0; d < DD; ++d) {
        float hv = h[r * DD + d];
        a += hv * ai[d];
        b += hv * aj[d];
    }
    for (int d = 0; d < DD; ++d) {
        float ev = emb[n * DD + d];
        a += ev * ai[DD + d];
        b += ev * aj[DD + d];
    }
    si[r] = a; sj[r] = b;
}

// ---------------------------------------------------------------------------
// GAT layer 1: exactly TOPK contiguous edges per destination node.
// block = one node (64 threads over D), softmax over 20 edges in shared mem.
// ---------------------------------------------------------------------------
__global__ void k_gat1(const float* __restrict__ h, const float* __restrict__ si,
                       const float* __restrict__ sj, const float* __restrict__ topv,
                       const int* __restrict__ topi, const float* __restrict__ bias1,
                       float* __restrict__ agg) {
    int r = blockIdx.x;
    int d = threadIdx.x;
    __shared__ float s_s[TOPK];
    __shared__ int   s_src[TOPK];
    int n = r & (NN - 1);
    int boff = r - n;                 // b*N
    if (d < TOPK) {
        int src = topi[n * TOPK + d] + boff;
        s_src[d] = src;
        s_s[d] = lrelu(si[r] + sj[src]) * topv[n * TOPK + d];
    }
    __syncthreads();
    float m = -3e38f;
    for (int k = 0; k < TOPK; ++k) m = fmaxf(m, s_s[k]);
    float den = 0.f;
    for (int k = 0; k < TOPK; ++k) den += __expf(s_s[k] - m);
    den = fmaxf(den, 1e-12f);
    float acc = 0.f;
    for (int k = 0; k < TOPK; ++k)
        acc += (__expf(s_s[k] - m) / den) * h[(long)s_src[k] * DD + d];
    agg[(long)r * DD + d] = acc + bias1[d];
}

// ---------------------------------------------------------------------------
// Column mean/var (population) over M rows of a M x 64 matrix.
// ---------------------------------------------------------------------------
__global__ void k_colstats(const float* __restrict__ X, int M,
                           float* __restrict__ mu, float* __restrict__ var) {
    int d = blockIdx.x, tid = threadIdx.x;
    float s = 0.f, s2 = 0.f;
    for (int r = tid; r < M; r += blockDim.x) {
        float v = X[(long)r * DD + d];
        s += v; s2 += v * v;
    }
    __shared__ float sh[256], sh2[256];
    sh[tid] = s; sh2[tid] = s2; __syncthreads();
    for (int st = 128; st > 0; st >>= 1) {
        if (tid < st) { sh[tid] += sh[tid + st]; sh2[tid] += sh2[tid + st]; }
        __syncthreads();
    }
    if (tid == 0) {
        float m = sh[0] / M;
        mu[d] = m;
        var[d] = sh2[0] / M - m * m;
    }
}

__global__ void k_bnrelu(const float* __restrict__ X, const float* __restrict__ mu,
                         const float* __restrict__ var, const float* __restrict__ g_,
                         const float* __restrict__ b_, float* __restrict__ out) {
    int gid = blockIdx.x * blockDim.x + threadIdx.x;
    if (gid >= BN * DD) return;
    int d = gid & 63;
    float v = g_[d] * (X[gid] - mu[d]) * rsqrtf(var[d] + EPS) + b_[d];
    out[gid] = fmaxf(v, 0.f);
}

// ---------------------------------------------------------------------------
// WMMA GEMM:  C[M x 64] = A[M x K] * B[K x 64]   (f32, V_WMMA_F32_16X16X4_F32)
// A addressed via COMPILE-TIME strides so the same kernel handles P^T
// (SAM=1, SAK=4096) with the address chains strength-reduced to pointer
// increments (no v_mul_u64 in the hot loop). One wave (32 threads) per
// 16-row tile; all four 16-col N-tiles per wave so A (and P, 64MB) is
// streamed from HBM exactly once.
// CDNA5 f32 16x16x4 lane layout: lane L holds A[L%16][2*(L/16)+v],
// B[2*(L/16)+v][L%16], and C/D[r+8*(L/16)][L%16].
// ---------------------------------------------------------------------------
template <long SAM, long SAK, int KK>
__global__ void k_wmma_gemm64(const float* __restrict__ A,
                              const float* __restrict__ B,
                              float* __restrict__ C) {
    int m0 = blockIdx.x * 16;
    int lane = threadIdx.x;           // 32 lanes, EXEC all ones
    int hl = lane >> 4;               // half-wave select (K offset 0 or 2)
    int lm = lane & 15;
    v8f acc0 = {0,0,0,0,0,0,0,0}, acc1 = acc0, acc2 = acc0, acc3 = acc0;
    const float* Ap = A + (long)(m0 + lm) * SAM + (long)(2 * hl) * SAK;
    const float* Bp = B + (long)(2 * hl) * DD + lm;
    #pragma unroll 4
    for (int k0 = 0; k0 < KK; k0 += 4) {
        if (k0 + 64 < KK) __builtin_prefetch(Ap + 64 * SAK, 0, 1);
        v2f a;
        a.x = Ap[0];
        a.y = Ap[SAK];
        v2f b0, b1, b2, b3;
        b0.x = Bp[0];  b0.y = Bp[DD];
        b1.x = Bp[16]; b1.y = Bp[DD + 16];
        b2.x = Bp[32]; b2.y = Bp[DD + 32];
        b3.x = Bp[48]; b3.y = Bp[DD + 48];
        acc0 = __builtin_amdgcn_wmma_f32_16x16x4_f32(false, a, false, b0, (short)0, acc0, false, false);
        acc1 = __builtin_amdgcn_wmma_f32_16x16x4_f32(false, a, false, b1, (short)0, acc1, false, false);
        acc2 = __builtin_amdgcn_wmma_f32_16x16x4_f32(false, a, false, b2, (short)0, acc2, false, false);
        acc3 = __builtin_amdgcn_wmma_f32_16x16x4_f32(false, a, false, b3, (short)0, acc3, false, false);
        Ap += 4 * SAK;
        Bp += 4 * DD;
    }
    for (int r = 0; r < 8; ++r) {
        int row = m0 + r + 8 * hl;
        float* Cr = C + (long)row * DD + lm;
        Cr[0]  = acc0[r];
        Cr[16] = acc1[r];
        Cr[32] = acc2[r];
        Cr[48] = acc3[r];
    }
}

// inverse norms of g rows (8192 x 64)
__global__ void k_invnorm(const float* __restrict__ g, float* __restrict__ invn) {
    int r = blockIdx.x * blockDim.x + threadIdx.x;
    if (r >= TWOBN) return;
    float ss = 0.f;
    const float* gr = g + (long)r * DD;
    for (int k = 0; k < DD; ++k) ss += gr[k] * gr[k];
    invn[r] = 1.0f / fmaxf(sqrtf(ss), 1e-12f);
}

// ---------------------------------------------------------------------------
// cos2 row + top-25: one block per row i (<BN). 8192 cosines staged in LDS
// (32KB — CDNA5 has 320KB/WGP), then 25 block-wide argmax passes with
// lax.top_k-compatible tie-breaking (smaller index wins on equal value).
// ---------------------------------------------------------------------------
__global__ void k_cos2_topk(const float* __restrict__ g, const float* __restrict__ invn,
                            float* __restrict__ topv2, int* __restrict__ topi2) {
    int i = blockIdx.x, tid = threadIdx.x;
    __shared__ float gi[DD];
    __shared__ float cosb[TWOBN];
    __shared__ float rv[256];
    __shared__ int   ri[256];
    if (tid < DD) gi[tid] = g[(long)i * DD + tid];
    __syncthreads();
    float inv_i = invn[i];
    for (int j = tid; j < TWOBN; j += 256) {
        const float* gr = g + (long)j * DD;
        float dot = 0.f;
        #pragma unroll
        for (int k = 0; k < DD; ++k) dot += gi[k] * gr[k];
        cosb[j] = dot * inv_i * invn[j];
    }
    __syncthreads();
    for (int it = 0; it < EXTRA; ++it) {
        float bv = -3e38f; int bi = 1 << 30;
        for (int j = tid; j < TWOBN; j += 256) {
            float v = cosb[j];
            if (v > bv || (v == bv && j < bi)) { bv = v; bi = j; }
        }
        rv[tid] = bv; ri[tid] = bi; __syncthreads();
        for (int s = 128; s > 0; s >>= 1) {
            if (tid < s) {
                if (rv[tid + s] > rv[tid] ||
                    (rv[tid + s] == rv[tid] && ri[tid + s] < ri[tid])) {
                    rv[tid] = rv[tid + s]; ri[tid] = ri[tid + s];
                }
            }
            __syncthreads();
        }
        if (tid == 0) {
            topv2[(long)i * EXTRA + it] = rv[0];
            topi2[(long)i * EXTRA + it] = ri[0];
            cosb[ri[0]] = -3e38f;
        }
        __syncthreads();
    }
}

// t2i/t2j = h2 @ att (per-row dots)
__global__ void k_t2(const float* __restrict__ h2, const float* __restrict__ ai,
                     const float* __restrict__ aj, float* __restrict__ t2i,
                     float* __restrict__ t2j) {
    int r = blockIdx.x * blockDim.x + threadIdx.x;
    if (r >= BN) return;
    float a = 0.f, b = 0.f;
    const float* hr = h2 + (long)r * DD;
    for (int d = 0; d < DD; ++d) { a += hr[d] * ai[d]; b += hr[d] * aj[d]; }
    t2i[r] = a; t2j[r] = b;
}

// segment softmax #2, irregular segments over dst2 (only dst2 < BN survive)
__global__ void k_e2_max(const float* __restrict__ t2i, const float* __restrict__ t2j,
                         const float* __restrict__ topv2, const int* __restrict__ topi2,
                         float* __restrict__ sraw, unsigned* __restrict__ segmax) {
    int e = blockIdx.x * blockDim.x + threadIdx.x;
    if (e >= BN * EXTRA) return;
    int src = e / EXTRA;
    int dst = topi2[e];
    if (dst >= BN) { sraw[e] = 0.f; return; }
    float s = lrelu(t2i[dst] + t2j[src]) * topv2[e];
    sraw[e] = s;
    atomicMax(&segmax[dst], fenc(s));
}

__global__ void k_e2_sum(const int* __restrict__ topi2, const unsigned* __restrict__ segmax,
                         float* __restrict__ eexp, float* __restrict__ segsum) {
    int e = blockIdx.x * blockDim.x + threadIdx.x;
    if (e >= BN * EXTRA) return;
    int dst = topi2[e];
    if (dst >= BN) return;
    float ev = __expf(eexp[e] - fdec(segmax[dst]));   // eexp currently holds raw s
    eexp[e] = ev;
    atomicAdd(&segsum[dst], ev);
}

__global__ void k_e2_agg(const int* __restrict__ topi2, const float* __restrict__ eexp,
                         const float* __restrict__ segsum, const float* __restrict__ h2,
                         float* __restrict__ agg2) {
    long gid = (long)blockIdx.x * blockDim.x + threadIdx.x;
    if (gid >= (long)BN * EXTRA * DD) return;
    int e = (int)(gid >> 6), d = (int)(gid & 63);
    int dst = topi2[e];
    if (dst >= BN) return;
    float alpha = eexp[e] / fmaxf(segsum[dst], 1e-12f);
    int src = e / EXTRA;
    atomicAdd(&agg2[(long)dst * DD + d], h2[(long)src * DD + d] * alpha);
}

// x3 = relu(agg2 + bias2) * emb
__global__ void k_x3(const float* __restrict__ agg2, const float* __restrict__ bias2,
                     const float* __restrict__ emb, float* __restrict__ x3) {
    int gid = blockIdx.x * blockDim.x + threadIdx.x;
    if (gid >= BN * DD) return;
    int r = gid >> 6, d = gid & 63, n = r & (NN - 1);
    x3[gid] = fmaxf(agg2[gid] + bias2[d], 0.f) * emb[n * DD + d];
}

// output head: per-channel BN + relu + Linear(D,1)
__global__ void k_out(const float* __restrict__ x3, const float* __restrict__ mu,
                      const float* __restrict__ var, const float* __restrict__ g_,
                      const float* __restrict__ b_, const float* __restrict__ Wout,
                      const float* __restrict__ bout, float* __restrict__ out) {
    int r = blockIdx.x * blockDim.x + threadIdx.x;
    if (r >= BN) return;
    float acc = bout[0];
    const float* xr = x3 + (long)r * DD;
    for (int d = 0; d < DD; ++d) {
        float y = g_[d] * (xr[d] - mu[d]) * rsqrtf(var[d] + EPS) + b_[d];
        acc += fmaxf(y, 0.f) * Wout[d];
    }
    out[r] = acc;
}

// ---------------------------------------------------------------------------
extern "C" void kernel_launch(void* const* d_in, const int* in_sizes, int n_in,
                              void* d_out, int out_size, void* d_ws, size_t ws_size,
                              hipStream_t stream) {
    const float* data   = (const float*)d_in[0];
    const float* emb    = (const float*)d_in[1];
    const float* W1     = (const float*)d_in[2];
    const float* att_i1 = (const float*)d_in[3];
    const float* att_j1 = (const float*)d_in[4];
    const float* bias1  = (const float*)d_in[5];
    const float* bn1_g  = (const float*)d_in[6];
    const float* bn1_b  = (const float*)d_in[7];
    const float* P      = (const float*)d_in[8];
    const float* W2     = (const float*)d_in[9];
    const float* att_i2 = (const float*)d_in[10];
    const float* att_j2 = (const float*)d_in[11];
    const float* bias2  = (const float*)d_in[12];
    const float* bnO_g  = (const float*)d_in[13];
    const float* bnO_b  = (const float*)d_in[14];
    const float* W_out  = (const float*)d_in[15];
    const float* b_out  = (const float*)d_in[16];
    float* out = (float*)d_out;

    char* w = (char*)d_ws;
    auto carve = [&](size_t nfloats) -> float* {
        float* p = (float*)w;
        w += nfloats * sizeof(float);
        return p;
    };
    float* topv   = carve(NN * TOPK);
    int*   topi   = (int*)carve(NN * TOPK);
    float* h      = carve((size_t)BN * DD);
    float* si     = carve(BN);
    float* sj     = carve(BN);
    float* agg    = carve((size_t)BN * DD);
    float* mu1    = carve(DD);
    float* var1   = carve(DD);
    float* g      = carve((size_t)TWOBN * DD);   // [gcn ; ef]
    float* invn2  = carve(TWOBN);
    float* topv2  = carve((size_t)BN * EXTRA);
    int*   topi2  = (int*)carve((size_t)BN * EXTRA);
    float* h2     = carve((size_t)BN * DD);
    float* t2i    = carve(BN);
    float* t2j    = carve(BN);
    float* eexp   = carve((size_t)BN * EXTRA);
    unsigned* segmax = (unsigned*)carve(BN);
    float* segsum = carve(BN);
    float* agg2   = carve((size_t)BN * DD);
    float* x3     = carve((size_t)BN * DD);
    float* mu2    = carve(DD);
    float* var2   = carve(DD);

    // Stage 1: emb cosine top-20
    k_topk_emb<<<1, NN, 0, stream>>>(emb, topv, topi);
    // h = x @ W1
    k_h<<<(BN * DD + 255) / 256, 256, 0, stream>>>(data, W1, h);
    // attention scalars
    k_sisj<<<(BN + 255) / 256, 256, 0, stream>>>(h, emb, att_i1, att_j1, si, sj);
    // GAT layer 1 (20 edges/node softmax + aggregate)
    k_gat1<<<BN, DD, 0, stream>>>(h, si, sj, topv, topi, bias1, agg);
    // BatchNorm1 + relu -> gcn (first half of g)
    k_colstats<<<DD, 256, 0, stream>>>(agg, BN, mu1, var1);
    k_bnrelu<<<(BN * DD + 255) / 256, 256, 0, stream>>>(agg, mu1, var1, bn1_g, bn1_b, g);
    // ef = P^T @ gcn via f32 WMMA (A strided as P^T: SAM=1, SAK=4096, K=4096)
    k_wmma_gemm64<1L, (long)BN, BN><<<BN / 16, 32, 0, stream>>>(P, g, g + (size_t)BN * DD);
    // cosine(g, g) rows<BN, top-25
    k_invnorm<<<(TWOBN + 255) / 256, 256, 0, stream>>>(g, invn2);
    k_cos2_topk<<<BN, 256, 0, stream>>>(g, invn2, topv2, topi2);
    // h2 = gcn @ W2 via WMMA (only first BN rows are ever consumed downstream)
    k_wmma_gemm64<(long)DD, 1L, DD><<<BN / 16, 32, 0, stream>>>(g, W2, h2);
    k_t2<<<(BN + 255) / 256, 256, 0, stream>>>(h2, att_i2, att_j2, t2i, t2j);
    // segment softmax #2 (irregular) + aggregation
    hipMemsetAsync(segmax, 0, BN * sizeof(unsigned), stream);
    hipMemsetAsync(segsum, 0, BN * sizeof(float), stream);
    hipMemsetAsync(agg2, 0, (size_t)BN * DD * sizeof(float), stream);
    k_e2_max<<<(BN * EXTRA + 255) / 256, 256, 0, stream>>>(t2i, t2j, topv2, topi2, eexp, segmax);
    k_e2_sum<<<(BN * EXTRA + 255) / 256, 256, 0, stream>>>(topi2, segmax, eexp, segsum);
    k_e2_agg<<<(int)(((long)BN * EXTRA * DD + 255) / 256), 256, 0, stream>>>(topi2, eexp, segsum, h2, agg2);
    // output head
    k_x3<<<(BN * DD + 255) / 256, 256, 0, stream>>>(agg2, bias2, emb, x3);
    k_colstats<<<DD, 256, 0, stream>>>(x3, BN, mu2, var2);
    k_out<<<(BN + 255) / 256, 256, 0, stream>>>(x3, mu2, var2, bnO_g, bnO_b, W_out, b_out, out);
}